// Rattle_53377853554894
// MI455X (gfx1250) — compile-verified
//
#include <hip/hip_runtime.h>

typedef __attribute__((ext_vector_type(2))) float v2f;
typedef __attribute__((ext_vector_type(8))) float v8f;

constexpr int kB = 64;          // replicas
constexpr int kC = 64;          // constraints
constexpr int kN = 4096;        // atoms
constexpr int kK = kN * 3;      // 12288 flattened reduction dim
constexpr int KC = 128;         // K staged per LDS tile
constexpr int JSTR = 132;       // padded LDS row stride (floats) -> conflict-free b64 frag loads
constexpr float kTol = 1e-6f;
constexpr int kIters = 4;

#define WMMA_F32(a, b, c) \
    __builtin_amdgcn_wmma_f32_16x16x4_f32(false, (a), false, (b), (short)0, (c), false, false)

// ---------------------------------------------------------------------------
// Kernel 1: split-K Gram matrix G[b,i,j] = sum_k (jac[i,k]/mas)*jac[j,k]
// plus b0 partial b0[b,i] = sum_k (jac[i,k]/mas)*mom[k].
// 64 threads (2 waves); each wave owns a 32x64 output block (8 f32 WMMA accs).
// Global->LDS staging is double-buffered via GLOBAL_LOAD_ASYNC_TO_LDS_B128.
// ---------------------------------------------------------------------------
__global__ __launch_bounds__(64)
void rattle_gram(const float* __restrict__ mom, const float* __restrict__ mas,
                 const float* __restrict__ jac, float* __restrict__ gpart,
                 float* __restrict__ b0part, int chunk)
{
    const int s = blockIdx.x;          // split index
    const int b = blockIdx.y;          // batch
    __shared__ __align__(16) float jl[2][kC * JSTR];  // 2 x (64 rows x 128, padded)
    __shared__ float im[2][KC];        // 1/mas per k
    __shared__ float wm[2][KC];        // mom[k]/mas per k

    const int t    = threadIdx.x;      // 0..63
    const int lane = t & 31;
    const int w    = t >> 5;           // wave 0/1 -> output rows [32w, 32w+32)
    const int rsel = lane & 15;
    const int kp   = (lane >> 4) << 1; // 0 or 2: K sub-offset within frag
    const int m0   = w * 32;
    const int m1   = m0 + 16;

    const float* jb = jac + (size_t)b * kC * kK;
    const int k0 = s * chunk;

    v8f a00 = {}, a01 = {}, a02 = {}, a03 = {};   // rows m0..m0+15, col tiles 0..3
    v8f a10 = {}, a11 = {}, a12 = {}, a13 = {};   // rows m1..m1+15, col tiles 0..3
    float b0acc = 0.0f;

    auto prefetch = [&](int kb, int nb) {
        // small per-k vectors (regular DS stores; barrier makes them visible)
        #pragma unroll
        for (int j = 0; j < 2; ++j) {
            const int idx = t + 64 * j;
            const int k = kb + idx;
            const float r = 1.0f / mas[b * kN + k / 3];
            im[nb][idx] = r;
            wm[nb][idx] = r * mom[(size_t)b * kK + k];
        }
        // 64 rows x 128 floats, async b128 per lane (ASYNCcnt-tracked)
        for (int i = t; i < kC * (KC / 4); i += 64) {
            const int row = i >> 5;    // 32 float4-segments per row
            const int seg = i & 31;
            const float* src = jb + (size_t)row * kK + kb + seg * 4;
            const unsigned dst = (unsigned)(size_t)(const void*)&jl[nb][row * JSTR + seg * 4];
            asm volatile("global_load_async_to_lds_b128 %0, %1, off"
                         :: "v"(dst), "v"(src) : "memory");
        }
    };

    prefetch(k0, 0);
    asm volatile("s_wait_asynccnt 0" ::: "memory");
    __syncthreads();

    int buf = 0;
    for (int st = 0; st < chunk; st += KC) {
        const int nb = buf ^ 1;
        if (st + KC < chunk) prefetch(k0 + st + KC, nb);   // overlap HBM with WMMA

        const float* J0 = &jl[buf][0];
        for (int kk = 0; kk < KC; kk += 4) {
            const int ko = kk + kp;
            const v2f sc = *(const v2f*)&im[buf][ko];
            v2f af0 = *(const v2f*)&J0[(m0 + rsel) * JSTR + ko] * sc;  // A = jac/mas
            v2f af1 = *(const v2f*)&J0[(m1 + rsel) * JSTR + ko] * sc;
            v2f bb;
            bb = *(const v2f*)&J0[(rsel) * JSTR + ko];
            a00 = WMMA_F32(af0, bb, a00);
            a10 = WMMA_F32(af1, bb, a10);
            bb = *(const v2f*)&J0[(16 + rsel) * JSTR + ko];
            a01 = WMMA_F32(af0, bb, a01);
            a11 = WMMA_F32(af1, bb, a11);
            bb = *(const v2f*)&J0[(32 + rsel) * JSTR + ko];
            a02 = WMMA_F32(af0, bb, a02);
            a12 = WMMA_F32(af1, bb, a12);
            bb = *(const v2f*)&J0[(48 + rsel) * JSTR + ko];
            a03 = WMMA_F32(af0, bb, a03);
            a13 = WMMA_F32(af1, bb, a13);
        }
        {   // b0 partial: thread t owns row t, full 128-k window
            const float* rj = &jl[buf][t * JSTR];
            const float* rw = &wm[buf][0];
            #pragma unroll 8
            for (int k = 0; k < KC; ++k) b0acc = fmaf(rj[k], rw[k], b0acc);
        }
        asm volatile("s_wait_asynccnt 0" ::: "memory");    // next tile landed
        __syncthreads();                                   // all waves done with 'buf'
        buf = nb;
    }

    // write G partial: C/D layout -> element v is row base+v+8*(lane>>4), col = tile*16+rsel
    const size_t gbase = ((size_t)s * kB + b) * (kC * kC);
    const int rhi = lane >> 4;
    #pragma unroll
    for (int v = 0; v < 8; ++v) {
        const int ma = m0 + v + 8 * rhi;
        const int mb = m1 + v + 8 * rhi;
        gpart[gbase + ma * kC +  0 + rsel] = a00[v];
        gpart[gbase + ma * kC + 16 + rsel] = a01[v];
        gpart[gbase + ma * kC + 32 + rsel] = a02[v];
        gpart[gbase + ma * kC + 48 + rsel] = a03[v];
        gpart[gbase + mb * kC +  0 + rsel] = a10[v];
        gpart[gbase + mb * kC + 16 + rsel] = a11[v];
        gpart[gbase + mb * kC + 32 + rsel] = a12[v];
        gpart[gbase + mb * kC + 48 + rsel] = a13[v];
    }
    b0part[((size_t)s * kB + b) * kC + t] = b0acc;
}

// ---------------------------------------------------------------------------
// Kernel 2: per-batch reduce partials, J = -dtm*G, LU w/ partial pivoting,
// 4 Newton iterations with convergence guard (matches jnp.linalg.solve path)
// ---------------------------------------------------------------------------
__global__ __launch_bounds__(64)
void rattle_solve(const float* __restrict__ gpart, const float* __restrict__ b0part,
                  const float* __restrict__ dtm, float* __restrict__ lmd_out, int nsplit)
{
    const int b = blockIdx.x;
    const int t = threadIdx.x;     // 64 threads, one matrix row each
    __shared__ float J[kC * 65];   // original J (padded)
    __shared__ float LU[kC * 65];  // LU factors (padded)
    __shared__ float b0[kC], lmd[kC], rhs[kC], red[kC];
    __shared__ int pivk[kC];
    __shared__ int psel;
    __shared__ int cflag;

    const float dt = dtm[b];
    for (int idx = t; idx < kC * kC; idx += 64) {
        float g = 0.0f;
        for (int s = 0; s < nsplit; ++s)
            g += gpart[((size_t)s * kB + b) * (kC * kC) + idx];
        const float jv = -dt * g;
        const int r = idx >> 6, c = idx & 63;
        J[r * 65 + c]  = jv;
        LU[r * 65 + c] = jv;
    }
    {
        float s0 = 0.0f;
        for (int s = 0; s < nsplit; ++s)
            s0 += b0part[((size_t)s * kB + b) * kC + t];
        b0[t]  = s0;
        lmd[t] = 0.0f;
    }
    __syncthreads();

    // LU factorization with partial pivoting
    for (int k = 0; k < kC; ++k) {
        if (t == 0) {
            int p = k; float best = fabsf(LU[k * 65 + k]);
            for (int i = k + 1; i < kC; ++i) {
                const float v = fabsf(LU[i * 65 + k]);
                if (v > best) { best = v; p = i; }
            }
            psel = p; pivk[k] = p;
        }
        __syncthreads();
        const int p = psel;
        if (p != k) {  // swap rows k<->p; thread t owns column t
            const float tmp = LU[k * 65 + t];
            LU[k * 65 + t] = LU[p * 65 + t];
            LU[p * 65 + t] = tmp;
        }
        __syncthreads();
        if (t > k) {
            const float l = LU[t * 65 + k] / LU[k * 65 + k];
            LU[t * 65 + k] = l;
            for (int j = k + 1; j < kC; ++j)
                LU[t * 65 + j] = fmaf(-l, LU[k * 65 + j], LU[t * 65 + j]);
        }
        __syncthreads();
    }

    for (int it = 0; it < kIters; ++it) {
        // con = b0 + J*lmd
        float c = b0[t];
        for (int j = 0; j < kC; ++j) c = fmaf(J[t * 65 + j], lmd[j], c);
        rhs[t] = c;
        red[t] = fabsf(c);
        __syncthreads();
        if (t == 0) {
            float mx = 0.0f;
            for (int i = 0; i < kC; ++i) mx = fmaxf(mx, red[i]);
            cflag = (mx < kTol) ? 1 : 0;
            for (int k = 0; k < kC; ++k) {   // apply row interchanges to rhs
                const int p = pivk[k];
                const float tmp = rhs[k]; rhs[k] = rhs[p]; rhs[p] = tmp;
            }
        }
        __syncthreads();
        // forward substitution (unit lower)
        for (int k = 0; k < kC; ++k) {
            if (t > k) rhs[t] = fmaf(-LU[t * 65 + k], rhs[k], rhs[t]);
            __syncthreads();
        }
        // back substitution
        for (int k = kC - 1; k >= 0; --k) {
            if (t == k) rhs[k] /= LU[k * 65 + k];
            __syncthreads();
            if (t < k) rhs[t] = fmaf(-LU[t * 65 + k], rhs[k], rhs[t]);
            __syncthreads();
        }
        if (!cflag) lmd[t] -= rhs[t];
        __syncthreads();
    }
    lmd_out[b * kC + t] = lmd[t];
}

// ---------------------------------------------------------------------------
// Kernel 3: mom_out = mom - dtm * sum_c lmd[c]*jac[c,:]  (pure bandwidth pass)
// ---------------------------------------------------------------------------
__global__ __launch_bounds__(256)
void rattle_apply(const float* __restrict__ mom, const float* __restrict__ jac,
                  const float* __restrict__ dtm, const float* __restrict__ lmd_in,
                  float* __restrict__ mom_out)
{
    const int b   = blockIdx.y;
    const int seg = blockIdx.x;            // 12 segments of 1024 floats
    const int t   = threadIdx.x;
    __shared__ float lmd[kC];
    if (t < kC) lmd[t] = lmd_in[b * kC + t];
    __syncthreads();
    const int kx = seg * 1024 + t * 4;
    const float* jb = jac + (size_t)b * kC * kK + kx;
    float4 s = make_float4(0.f, 0.f, 0.f, 0.f);
    #pragma unroll 4
    for (int c = 0; c < kC; ++c) {
        const float4 jv = *(const float4*)(jb + (size_t)c * kK);
        const float lc = lmd[c];
        s.x = fmaf(lc, jv.x, s.x);
        s.y = fmaf(lc, jv.y, s.y);
        s.z = fmaf(lc, jv.z, s.z);
        s.w = fmaf(lc, jv.w, s.w);
    }
    const float4 mv = *(const float4*)(mom + (size_t)b * kK + kx);
    const float dt = dtm[b];
    float4 o;
    o.x = mv.x - dt * s.x;
    o.y = mv.y - dt * s.y;
    o.z = mv.z - dt * s.z;
    o.w = mv.w - dt * s.w;
    *(float4*)(mom_out + (size_t)b * kK + kx) = o;
}

// ---------------------------------------------------------------------------
extern "C" void kernel_launch(void* const* d_in, const int* in_sizes, int n_in,
                              void* d_out, int out_size, void* d_ws, size_t ws_size,
                              hipStream_t stream)
{
    const float* mom = (const float*)d_in[0];
    const float* mas = (const float*)d_in[1];
    const float* dtm = (const float*)d_in[2];
    const float* jac = (const float*)d_in[3];
    float* out = (float*)d_out;
    float* lmd_out = out + (size_t)kB * kK;    // outputs: [mom_out | lmd]

    // Split-K count chosen deterministically from ws_size (same every call).
    const size_t per_split = (size_t)kB * kC * kC * sizeof(float)
                           + (size_t)kB * kC * sizeof(float);
    int S = 8;                                  // S in {8,4,2,1}: chunk stays a multiple of 128
    while (S > 1 && (size_t)S * per_split > ws_size) S >>= 1;
    float* gpart  = (float*)d_ws;
    float* b0part = gpart + (size_t)S * kB * kC * kC;
    const int chunk = kK / S;

    dim3 g1(S, kB);
    rattle_gram<<<g1, 64, 0, stream>>>(mom, mas, jac, gpart, b0part, chunk);
    rattle_solve<<<kB, 64, 0, stream>>>(gpart, b0part, dtm, lmd_out, S);
    dim3 g3(kK / 1024, kB);
    rattle_apply<<<g3, 256, 0, stream>>>(mom, jac, dtm, lmd_out, out);
}